// Model_Q50_4758823764471
// MI455X (gfx1250) — compile-verified
//
#include <hip/hip_runtime.h>
#include <stdint.h>
#include <math.h>

// ---------------------------------------------------------------------------
// Types for CDNA5 WMMA f32 16x16x4
// ---------------------------------------------------------------------------
typedef float v2f __attribute__((ext_vector_type(2)));
typedef float v8f __attribute__((ext_vector_type(8)));

#define FB_BETA 1
#define FB_GELU 2
#define FB_BIAS 4

// ---------------------------------------------------------------------------
// threefry2x32 (JAX-compatible), usable host + device
// ---------------------------------------------------------------------------
__host__ __device__ static inline void threefry2x32(uint32_t k0, uint32_t k1,
                                                    uint32_t x0, uint32_t x1,
                                                    uint32_t* o0, uint32_t* o1) {
  uint32_t ks[3];
  ks[0] = k0; ks[1] = k1; ks[2] = k0 ^ k1 ^ 0x1BD11BDAu;
  const int rot[2][4] = {{13, 15, 26, 6}, {17, 29, 16, 24}};
  x0 += ks[0]; x1 += ks[1];
  for (int i = 0; i < 5; ++i) {
    const int* r = rot[i & 1];
    for (int j = 0; j < 4; ++j) {
      x0 += x1;
      x1 = (x1 << r[j]) | (x1 >> (32 - r[j]));
      x1 ^= x0;
    }
    x0 += ks[(i + 1) % 3];
    x1 += ks[(i + 2) % 3] + (uint32_t)(i + 1);
  }
  *o0 = x0; *o1 = x1;
}

__device__ __forceinline__ float gelu_f(float x) {
  return 0.5f * x * (1.0f + erff(x * 0.7071067811865475f));
}
__device__ __forceinline__ float wred_max(float v) {
  for (int o = 16; o > 0; o >>= 1) v = fmaxf(v, __shfl_xor(v, o, 32));
  return v;
}
__device__ __forceinline__ float wred_sum(float v) {
  for (int o = 16; o > 0; o >>= 1) v += __shfl_xor(v, o, 32);
  return v;
}

// ---------------------------------------------------------------------------
// Generic WMMA f32 GEMM: C[M,N] = act(A'[M,K] @ B[K,N] + bias (+C if beta))
// A' = A with optional circular row shift within sequences of length seqL.
// Block: 256 threads (8 waves), 128x64 output tile, BK=16, double-buffered LDS.
// Each wave: 32x32 output = four 16x16 WMMA tiles (A/B fragment reuse:
// 2 LDS dwords per WMMA instead of 4).
// Requires K % 16 == 0 (true for all uses: 32, 512, 2048).
// ---------------------------------------------------------------------------
__global__ __launch_bounds__(256) void k_gemm(
    const float* __restrict__ A, const float* __restrict__ B,
    const float* __restrict__ bias, float* __restrict__ C,
    int M, int N, int K, int shiftS, int seqL, int flags) {
  __shared__ float As[2][128][17];
  __shared__ float Bs[2][16][65];
  const int tid = threadIdx.x;
  const int bm = blockIdx.y * 128, bn0 = blockIdx.x * 64;
  const int lane = tid & 31, wid = tid >> 5;
  const int rowbase = (wid >> 1) * 32;
  const int col0 = (wid & 1) * 32, col1 = col0 + 16;
  const int ln = lane & 15;
  const int m8 = (lane < 16) ? 0 : 8;
  const int klo = (lane < 16) ? 0 : 2;
  const bool n4 = ((N & 3) == 0);

  v8f acc00 = {0.f, 0.f, 0.f, 0.f, 0.f, 0.f, 0.f, 0.f};
  v8f acc01 = acc00, acc10 = acc00, acc11 = acc00;

  auto preload = [&](v8f& acc, int rb, int cb) {
    for (int v = 0; v < 8; ++v) {
      int r = bm + rb + v + m8;
      int c = bn0 + cb + ln;
      if (r < M && c < N) acc[v] = C[(size_t)r * N + c];
    }
  };
  if (flags & FB_BETA) {
    preload(acc00, rowbase, col0);
    preload(acc01, rowbase, col1);
    preload(acc10, rowbase + 16, col0);
    preload(acc11, rowbase + 16, col1);
  }

  auto load_tile = [&](int kt, int buf) {
    const int k0 = kt << 4;
    // A: 128 x 16, float4 per thread x2
    for (int i = tid; i < 512; i += 256) {
      int r = i >> 2, kq = (i & 3) << 2;
      int gr = bm + r;
      if (gr < M) {
        int sr = gr;
        if (seqL > 0) {
          int s = gr / seqL, l = gr % seqL;
          l += shiftS;
          if (l < 0) l += seqL;
          if (l >= seqL) l -= seqL;
          sr = s * seqL + l;
        }
        const float4 v = *(const float4*)(A + (size_t)sr * K + k0 + kq);
        As[buf][r][kq + 0] = v.x;
        As[buf][r][kq + 1] = v.y;
        As[buf][r][kq + 2] = v.z;
        As[buf][r][kq + 3] = v.w;
      } else {
        As[buf][r][kq + 0] = 0.f;
        As[buf][r][kq + 1] = 0.f;
        As[buf][r][kq + 2] = 0.f;
        As[buf][r][kq + 3] = 0.f;
      }
    }
    // B: 16 x 64, one float4 per thread
    {
      int kk = tid >> 4, c4 = (tid & 15) << 2;
      int gc = bn0 + c4;
      const float* bp = B + (size_t)(k0 + kk) * N;
      if (n4 && gc + 3 < N) {
        const float4 v = *(const float4*)(bp + gc);
        Bs[buf][kk][c4 + 0] = v.x;
        Bs[buf][kk][c4 + 1] = v.y;
        Bs[buf][kk][c4 + 2] = v.z;
        Bs[buf][kk][c4 + 3] = v.w;
      } else {
        for (int q = 0; q < 4; ++q)
          Bs[buf][kk][c4 + q] = (gc + q < N) ? bp[gc + q] : 0.f;
      }
    }
  };

  const int nT = K >> 4;
  load_tile(0, 0);
  __syncthreads();
  for (int kt = 0; kt < nT; ++kt) {
    const int cur = kt & 1;
    if (kt + 1 < nT) load_tile(kt + 1, cur ^ 1);
#pragma unroll
    for (int kk = 0; kk < 16; kk += 4) {
      v2f a0, a1, b0, b1;
      a0.x = As[cur][rowbase + ln][kk + klo];
      a0.y = As[cur][rowbase + ln][kk + klo + 1];
      a1.x = As[cur][rowbase + 16 + ln][kk + klo];
      a1.y = As[cur][rowbase + 16 + ln][kk + klo + 1];
      b0.x = Bs[cur][kk + klo][col0 + ln];
      b0.y = Bs[cur][kk + klo + 1][col0 + ln];
      b1.x = Bs[cur][kk + klo][col1 + ln];
      b1.y = Bs[cur][kk + klo + 1][col1 + ln];
      acc00 = __builtin_amdgcn_wmma_f32_16x16x4_f32(false, a0, false, b0,
                                                    (short)0, acc00, false, false);
      acc01 = __builtin_amdgcn_wmma_f32_16x16x4_f32(false, a0, false, b1,
                                                    (short)0, acc01, false, false);
      acc10 = __builtin_amdgcn_wmma_f32_16x16x4_f32(false, a1, false, b0,
                                                    (short)0, acc10, false, false);
      acc11 = __builtin_amdgcn_wmma_f32_16x16x4_f32(false, a1, false, b1,
                                                    (short)0, acc11, false, false);
    }
    __syncthreads();
  }

  auto store_tile = [&](v8f& acc, int rb, int cb) {
    for (int v = 0; v < 8; ++v) {
      int r = bm + rb + v + m8;
      if (r >= M) continue;
      int c = bn0 + cb + ln;
      if (c < N) {
        float val = acc[v];
        if (flags & FB_BIAS) val += bias[c];
        if (flags & FB_GELU) val = gelu_f(val);
        C[(size_t)r * N + c] = val;
      }
    }
  };
  store_tile(acc00, rowbase, col0);
  store_tile(acc01, rowbase, col1);
  store_tile(acc10, rowbase + 16, col0);
  store_tile(acc11, rowbase + 16, col1);
}

// ---------------------------------------------------------------------------
// data_embedding: circ conv(ENC_IN=1) + pos-embed + mark@tw + tb
// rows ordered bn = b*NODE + node (x), mark row = bn % B_ORIG (ref semantics)
// ---------------------------------------------------------------------------
__global__ void k_embed(const float* __restrict__ x, const float* __restrict__ xmark,
                        const float* __restrict__ convW, const float* __restrict__ tw,
                        const float* __restrict__ tb, float* __restrict__ out,
                        int Bo, int Node, int L, int Mk, int total) {
  int i = blockIdx.x * blockDim.x + threadIdx.x;
  if (i >= total) return;
  int d = i & 511;
  int r = i >> 9;
  int l = r % L, bn = r / L;
  int b = bn / Node, node = bn % Node;
  int lm = (l == 0) ? L - 1 : l - 1;
  int lp = (l == L - 1) ? 0 : l + 1;
  float xm1 = x[((size_t)(b * L + lm)) * Node + node];
  float x0c = x[((size_t)(b * L + l)) * Node + node];
  float xp1 = x[((size_t)(b * L + lp)) * Node + node];
  float v = xm1 * convW[d] + x0c * convW[512 + d] + xp1 * convW[1024 + d];
  int j2 = d & ~1;
  float div = expf((float)j2 * (-9.210340371976184f / 512.0f));
  float ang = (float)l * div;
  v += (d & 1) ? cosf(ang) : sinf(ang);
  int mb = bn % Bo;
  const float* mrow = xmark + ((size_t)mb * L + l) * Mk;
  for (int m = 0; m < Mk; ++m) v += mrow[m] * tw[m * 512 + d];
  v += tb[d];
  out[i] = v;
}

// out = LN(x (+ y)) * g + b  over rows of width 512 (in-place on x is safe)
__global__ __launch_bounds__(256) void k_add_ln(
    const float* __restrict__ x, const float* __restrict__ y,
    const float* __restrict__ g, const float* __restrict__ bb,
    float* __restrict__ out, int R) {
  __shared__ float red[256];
  int r = blockIdx.x;
  if (r >= R) return;
  int t = threadIdx.x;
  const float* xr = x + (size_t)r * 512;
  const float* yr = y ? (y + (size_t)r * 512) : nullptr;
  float v0 = xr[t] + (yr ? yr[t] : 0.f);
  float v1 = xr[t + 256] + (yr ? yr[t + 256] : 0.f);
  red[t] = v0 + v1;
  __syncthreads();
  for (int o = 128; o > 0; o >>= 1) { if (t < o) red[t] += red[t + o]; __syncthreads(); }
  float mean = red[0] * (1.f / 512.f);
  __syncthreads();
  float d0 = v0 - mean, d1 = v1 - mean;
  red[t] = d0 * d0 + d1 * d1;
  __syncthreads();
  for (int o = 128; o > 0; o >>= 1) { if (t < o) red[t] += red[t + o]; __syncthreads(); }
  float rstd = rsqrtf(red[0] * (1.f / 512.f) + 1e-5f);
  out[(size_t)r * 512 + t] = d0 * rstd * g[t] + bb[t];
  out[(size_t)r * 512 + t + 256] = d1 * rstd * g[t + 256] + bb[t + 256];
}

// y = elu(y * bng / sqrt(1+1e-5) + bnb)
__global__ void k_bn_elu(float* y, const float* __restrict__ g,
                         const float* __restrict__ b, int total) {
  int i = blockIdx.x * blockDim.x + threadIdx.x;
  if (i >= total) return;
  int d = i & 511;
  float v = y[i] * g[d] * 0.99999500003749969f + b[d];
  y[i] = v > 0.f ? v : expm1f(v);
}

// maxpool window 3 stride 2 pad 1 along L
__global__ void k_maxpool(const float* __restrict__ in, float* __restrict__ out,
                          int L, int Lout, int total) {
  int i = blockIdx.x * blockDim.x + threadIdx.x;
  if (i >= total) return;
  int d = i & 511;
  int r = i >> 9;
  int lo = r % Lout, bn = r / Lout;
  size_t base = (size_t)bn * L * 512;
  float m = -INFINITY;
  for (int dl = -1; dl <= 1; ++dl) {
    int li = 2 * lo + dl;
    if (li >= 0 && li < L) m = fmaxf(m, in[base + (size_t)li * 512 + d]);
  }
  out[i] = m;
}

// JAX randint(key,(L,U),0,Lk) via threefry counter mode; n = L*U (even here)
__global__ void k_fill_idx(int* __restrict__ idx, int n, int Lk,
                           uint32_t k0, uint32_t k1) {
  int j = blockIdx.x * blockDim.x + threadIdx.x;
  if (j >= n) return;
  int h = n >> 1;
  uint32_t t = (j < h) ? (uint32_t)j : (uint32_t)(j - h);
  uint32_t o0, o1;
  threefry2x32(k0, k1, t, (uint32_t)h + t, &o0, &o1);
  uint32_t bits = (j < h) ? o0 : o1;
  idx[j] = (int)(bits % (uint32_t)Lk);
}

// M[bh,l] = max_s q·k_idx - (1/Lk) sum_s q·k_idx   (Lk == L)
__global__ void k_meas(const float* __restrict__ Q, const float* __restrict__ Kb,
                       const int* __restrict__ idx, float* __restrict__ Mout,
                       int L, int U) {
  int bh = blockIdx.x;
  int h = bh & 7, bn = bh >> 3;
  int l = threadIdx.x;
  if (l >= L) return;
  const float* q = Q + ((size_t)(bn * L + l)) * 512 + h * 64;
  float mx = -INFINITY, sm = 0.f;
  for (int s = 0; s < U; ++s) {
    int kr = idx[l * U + s];
    const float* kp = Kb + ((size_t)(bn * L + kr)) * 512 + h * 64;
    float d = 0.f;
    for (int e = 0; e < 64; ++e) d += q[e] * kp[e];
    mx = fmaxf(mx, d);
    sm += d;
  }
  Mout[(size_t)bh * L + l] = mx - sm / (float)L;
}

// stable top-u selection (descending, ties -> lower index) per (b,h)
__global__ void k_topk(const float* __restrict__ Mb, int* __restrict__ top,
                       int L, int u) {
  if (threadIdx.x != 0) return;
  int bh = blockIdx.x;
  float vals[192];
  const float* mr = Mb + (size_t)bh * L;
  for (int i = 0; i < L; ++i) vals[i] = mr[i];
  for (int t = 0; t < u; ++t) {
    int bi = 0; float bv = vals[0];
    for (int i = 1; i < L; ++i)
      if (vals[i] > bv) { bv = vals[i]; bi = i; }
    top[(size_t)bh * u + t] = bi;
    vals[bi] = -INFINITY;
  }
}

__device__ __forceinline__ size_t attn_off(int bn, int h, int l, int d, int L, int mix) {
  return mix ? ((size_t)bn * L * 512 + ((size_t)h * L + l) * 64 + d)
             : (((size_t)(bn * L + l)) * 512 + h * 64 + d);
}

// ctx base (non-masked): broadcast V mean over L
__global__ void k_vmean_bcast(const float* __restrict__ V, float* __restrict__ out,
                              int L, int mix) {
  int bh = blockIdx.x;
  int h = bh & 7, bn = bh >> 3;
  int d = threadIdx.x;  // 64
  float s = 0.f;
  for (int l = 0; l < L; ++l) s += V[((size_t)(bn * L + l)) * 512 + h * 64 + d];
  s /= (float)L;
  for (int l = 0; l < L; ++l) out[attn_off(bn, h, l, d, L, mix)] = s;
}

// ctx base (masked): inclusive cumsum of V over L
__global__ void k_cumsum(const float* __restrict__ V, float* __restrict__ out,
                         int L, int mix, int total) {
  int i = blockIdx.x * blockDim.x + threadIdx.x;
  if (i >= total) return;
  int d = i & 63, h = (i >> 6) & 7, bn = i >> 9;
  float s = 0.f;
  for (int l = 0; l < L; ++l) {
    s += V[((size_t)(bn * L + l)) * 512 + h * 64 + d];
    out[attn_off(bn, h, l, d, L, mix)] = s;
  }
}

// overwrite top-u rows with softmax(QrK^T/8 [masked]) @ V ; wave per row
__global__ __launch_bounds__(128) void k_topattn(
    const float* __restrict__ Q, const float* __restrict__ Kb,
    const float* __restrict__ V, const int* __restrict__ top,
    float* __restrict__ out, int L, int u, int masked, int mix) {
  __shared__ float probs[4][192];
  int wid = threadIdx.x >> 5, lane = threadIdx.x & 31;
  int bh = blockIdx.x;
  int h = bh & 7, bn = bh >> 3;
  int t = blockIdx.y * 4 + wid;
  int lrow = (t < u) ? top[(size_t)bh * u + t] : 0;
  const float* q = Q + ((size_t)(bn * L + lrow)) * 512 + h * 64;
  float mx = -INFINITY;
  for (int s = lane; s < L; s += 32) {
    const float* kp = Kb + ((size_t)(bn * L + s)) * 512 + h * 64;
    float d = 0.f;
    for (int e = 0; e < 64; ++e) d += q[e] * kp[e];
    d *= 0.125f;
    if (masked && s > lrow) d = -INFINITY;
    probs[wid][s] = d;
    mx = fmaxf(mx, d);
  }
  mx = wred_max(mx);
  float sm = 0.f;
  for (int s = lane; s < L; s += 32) {
    float p = expf(probs[wid][s] - mx);
    probs[wid][s] = p;
    sm += p;
  }
  sm = wred_sum(sm);
  float inv = 1.f / sm;
  for (int s = lane; s < L; s += 32) probs[wid][s] *= inv;
  __syncthreads();
  for (int d = lane; d < 64; d += 32) {
    float a = 0.f;
    for (int s = 0; s < L; ++s)
      a += probs[wid][s] * V[((size_t)(bn * L + s)) * 512 + h * 64 + d];
    if (t < u) out[attn_off(bn, h, lrow, d, L, mix)] = a;
  }
}

// full attention (decoder cross): wave per (bn,h,l) row; S = encoder length
__global__ __launch_bounds__(128) void k_cross(
    const float* __restrict__ Q, const float* __restrict__ Kb,
    const float* __restrict__ V, float* __restrict__ out, int Lq, int S) {
  __shared__ float probs[4][64];
  int wid = threadIdx.x >> 5, lane = threadIdx.x & 31;
  int gw = blockIdx.x * 4 + wid;
  int total = 128 * 8 * Lq;
  bool act = gw < total;
  int l = act ? gw % Lq : 0;
  int bh = act ? gw / Lq : 0;
  int h = bh & 7, bn = bh >> 3;
  const float* q = Q + ((size_t)(bn * Lq + l)) * 512 + h * 64;
  float mx = -INFINITY;
  for (int s = lane; s < S; s += 32) {
    const float* kp = Kb + ((size_t)(bn * S + s)) * 512 + h * 64;
    float d = 0.f;
    for (int e = 0; e < 64; ++e) d += q[e] * kp[e];
    d *= 0.125f;
    probs[wid][s] = d;
    mx = fmaxf(mx, d);
  }
  mx = wred_max(mx);
  float sm = 0.f;
  for (int s = lane; s < S; s += 32) {
    float p = expf(probs[wid][s] - mx);
    probs[wid][s] = p;
    sm += p;
  }
  sm = wred_sum(sm);
  float inv = 1.f / sm;
  for (int s = lane; s < S; s += 32) probs[wid][s] *= inv;
  __syncthreads();
  for (int d = lane; d < 64; d += 32) {
    float a = 0.f;
    for (int s = 0; s < S; ++s)
      a += probs[wid][s] * V[((size_t)(bn * S + s)) * 512 + h * 64 + d];
    if (act) out[((size_t)(bn * Lq + l)) * 512 + h * 64 + d] = a;
  }
}

// wavg[k] = mean_j proj_w[k,j];  wavg[512] = mean(proj_b)
__global__ void k_wavg(const float* __restrict__ pw, const float* __restrict__ pb,
                       float* __restrict__ wavg) {
  int k = blockIdx.x * blockDim.x + threadIdx.x;
  if (k < 512) {
    float s = 0.f;
    for (int j = 0; j < 7; ++j) s += pw[k * 7 + j];
    wavg[k] = s * (1.f / 7.f);
  }
  if (k == 0) {
    float s = 0.f;
    for (int j = 0; j < 7; ++j) s += pb[j];
    wavg[512] = s * (1.f / 7.f);
  }
}

// dmean[r] = dot(X[r], wavg) + wavg[512]; wave per row
__global__ __launch_bounds__(256) void k_projmean(const float* __restrict__ X,
                                                  const float* __restrict__ wavg,
                                                  float* __restrict__ out, int R) {
  int wid = threadIdx.x >> 5, lane = threadIdx.x & 31;
  int r = blockIdx.x * 8 + wid;
  if (r >= R) return;
  const float* xr = X + (size_t)r * 512;
  float s = 0.f;
  for (int k = lane; k < 512; k += 32) s += xr[k] * wavg[k];
  s = wred_sum(s);
  if (lane == 0) out[r] = s + wavg[512];
}

// y[b,t,node] = dmean[(b*32+node)*144 + 96 + t]
__global__ void k_ygather(const float* __restrict__ dmean, float* __restrict__ y,
                          int total) {
  int i = blockIdx.x * blockDim.x + threadIdx.x;
  if (i >= total) return;
  int node = i & 31;
  int t = (i >> 5) % 48;
  int b = i / (48 * 32);
  y[i] = dmean[(size_t)(b * 32 + node) * 144 + 96 + t];
}

// ---------------------------------------------------------------------------
// Host orchestration
// ---------------------------------------------------------------------------
namespace {
struct PW {
  void* const* din;
  int i;
  const float* n() { return (const float*)din[i++]; }
};
struct AttnP { const float *wq, *bq, *wk, *bk, *wv, *bv, *wo, *bo; };
static AttnP rdAttn(PW& p) {
  AttnP a;
  a.wq = p.n(); a.bq = p.n(); a.wk = p.n(); a.bk = p.n();
  a.wv = p.n(); a.bv = p.n(); a.wo = p.n(); a.bo = p.n();
  return a;
}
struct EncLP { AttnP a; const float *w1, *b1, *w2, *b2, *l1g, *l1b, *l2g, *l2b; };
struct ConvLP { const float *w, *b, *bng, *bnb; };
struct DecLP {
  AttnP sa, ca;
  const float *w1, *b1, *w2, *b2, *l1g, *l1b, *l2g, *l2b, *l3g, *l3b;
};

static void gemm(hipStream_t st, const float* A, const float* B, const float* bias,
                 float* C, int M, int N, int K, int flags, int shiftS = 0,
                 int seqL = 0) {
  dim3 g((N + 63) / 64, (M + 127) / 128), b(256);
  k_gemm<<<g, b, 0, st>>>(A, B, bias, C, M, N, K, shiftS, seqL, flags);
}
static inline int gs(int total) { return (total + 255) / 256; }
}  // namespace

extern "C" void kernel_launch(void* const* d_in, const int* in_sizes, int n_in,
                              void* d_out, int out_size, void* d_ws, size_t ws_size,
                              hipStream_t stream) {
  (void)in_sizes; (void)n_in; (void)out_size; (void)ws_size;
  const float* x_enc = (const float*)d_in[0];       // (4,192,32,1)
  const float* x_mark_enc = (const float*)d_in[1];  // (4,192,4)
  const float* x_dec = (const float*)d_in[2];       // (4,144,32,1)
  const float* x_mark_dec = (const float*)d_in[3];  // (4,144,4)

  PW pw{d_in, 4};
  const float *ee_conv = pw.n(), *ee_tw = pw.n(), *ee_tb = pw.n();
  const float *de_conv = pw.n(), *de_tw = pw.n(), *de_tb = pw.n();
  EncLP enc[3];
  for (int i = 0; i < 3; ++i) {
    enc[i].a = rdAttn(pw);
    enc[i].w1 = pw.n(); enc[i].b1 = pw.n(); enc[i].w2 = pw.n(); enc[i].b2 = pw.n();
    enc[i].l1g = pw.n(); enc[i].l1b = pw.n(); enc[i].l2g = pw.n(); enc[i].l2b = pw.n();
  }
  ConvLP cvl[2];
  for (int i = 0; i < 2; ++i) {
    cvl[i].w = pw.n(); cvl[i].b = pw.n(); cvl[i].bng = pw.n(); cvl[i].bnb = pw.n();
  }
  const float *enc_ng = pw.n(), *enc_nb = pw.n();
  DecLP dec[2];
  for (int i = 0; i < 2; ++i) {
    dec[i].sa = rdAttn(pw);
    dec[i].ca = rdAttn(pw);
    dec[i].w1 = pw.n(); dec[i].b1 = pw.n(); dec[i].w2 = pw.n(); dec[i].b2 = pw.n();
    dec[i].l1g = pw.n(); dec[i].l1b = pw.n(); dec[i].l2g = pw.n(); dec[i].l2b = pw.n();
    dec[i].l3g = pw.n(); dec[i].l3b = pw.n();
  }
  const float *dec_ng = pw.n(), *dec_nb = pw.n();
  const float *proj_w = pw.n(), *proj_b = pw.n();
  const float *fl_w1 = pw.n(), *fl_b1 = pw.n(), *fl_w2 = pw.n(), *fl_b2 = pw.n();

  // ---- workspace layout (floats) ----
  float* ws = (float*)d_ws;
  const size_t SZ = (size_t)24576 * 512;
  float* Xb = ws;
  float* Yb = ws + SZ;
  float* Qb = ws + 2 * SZ;
  float* Kbuf = ws + 3 * SZ;
  float* Vbuf = ws + 4 * SZ;
  float* Tb = ws + 5 * SZ;
  float* Hb = ws + 6 * SZ;                    // 24576*2048
  float* Eb = Hb + (size_t)24576 * 2048;      // 128*48*512
  float* Mb = Eb + (size_t)128 * 48 * 512;    // 128*8*192
  int* TopI = (int*)(Mb + 128 * 8 * 192);     // 128*8*30
  int* IdxI = TopI + 128 * 8 * 30;            // 192*30
  float* Dm = (float*)(IdxI + 192 * 30);      // 128*144
  float* Yg = Dm + 128 * 144;                 // 4*48*32
  float* Wavg = Yg + 4 * 48 * 32;             // 513

  const int BN = 128;

  // ========================= ENCODER =========================
  int L = 192, R = BN * L;
  k_embed<<<gs(R * 512), 256, 0, stream>>>(x_enc, x_mark_enc, ee_conv, ee_tw, ee_tb,
                                           Xb, 4, 32, L, 4, R * 512);
  for (int i = 0; i < 3; ++i) {
    const EncLP& p = enc[i];
    gemm(stream, Xb, p.a.wq, p.a.bq, Qb, R, 512, 512, FB_BIAS);
    gemm(stream, Xb, p.a.wk, p.a.bk, Kbuf, R, 512, 512, FB_BIAS);
    gemm(stream, Xb, p.a.wv, p.a.bv, Vbuf, R, 512, 512, FB_BIAS);
    int U = (L == 192) ? 30 : (L == 96) ? 25 : 20;  // 5*ceil(ln L)
    int u = U;
    uint32_t fk0, fk1;
    threefry2x32(0u, 1234u, 0u, (uint32_t)i, &fk0, &fk1);  // fold_in(key(1234), i)
    k_fill_idx<<<gs(L * U), 256, 0, stream>>>(IdxI, L * U, L, fk0, fk1);
    k_meas<<<1024, L, 0, stream>>>(Qb, Kbuf, IdxI, Mb, L, U);
    k_topk<<<1024, 32, 0, stream>>>(Mb, TopI, L, u);
    k_vmean_bcast<<<1024, 64, 0, stream>>>(Vbuf, Tb, L, 0);
    k_topattn<<<dim3(1024, (u + 3) / 4), 128, 0, stream>>>(Qb, Kbuf, Vbuf, TopI, Tb,
                                                           L, u, 0, 0);
    gemm(stream, Tb, p.a.wo, p.a.bo, Yb, R, 512, 512, FB_BIAS);
    k_add_ln<<<R, 256, 0, stream>>>(Xb, Yb, p.l1g, p.l1b, Xb, R);
    gemm(stream, Xb, p.w1, p.b1, Hb, R, 2048, 512, FB_BIAS | FB_GELU);
    gemm(stream, Hb, p.w2, p.b2, Yb, R, 512, 2048, FB_BIAS);
    k_add_ln<<<R, 256, 0, stream>>>(Xb, Yb, p.l2g, p.l2b, Xb, R);
    if (i < 2) {
      const ConvLP& c = cvl[i];
      gemm(stream, Xb, c.w, c.b, Tb, R, 512, 512, FB_BIAS, -1, L);
      gemm(stream, Xb, c.w + 512 * 512, nullptr, Tb, R, 512, 512, FB_BETA, 0, L);
      gemm(stream, Xb, c.w + 2 * 512 * 512, nullptr, Tb, R, 512, 512, FB_BETA, 1, L);
      k_bn_elu<<<gs(R * 512), 256, 0, stream>>>(Tb, c.bng, c.bnb, R * 512);
      int Lout = (L - 1) / 2 + 1;  // 192->96, 96->48
      k_maxpool<<<gs(BN * Lout * 512), 256, 0, stream>>>(Tb, Xb, L, Lout,
                                                         BN * Lout * 512);
      L = Lout; R = BN * L;
    }
  }
  // enc_norm -> encoder memory (128*48,512)
  k_add_ln<<<R, 256, 0, stream>>>(Xb, nullptr, enc_ng, enc_nb, Eb, R);
  const int Renc = R;  // 128*48

  // ========================= DECODER =========================
  const int Ld = 144, Rd = BN * Ld;
  k_embed<<<gs(Rd * 512), 256, 0, stream>>>(x_dec, x_mark_dec, de_conv, de_tw, de_tb,
                                            Xb, 4, 32, Ld, 4, Rd * 512);
  for (int i = 0; i < 2; ++i) {
    const DecLP& p = dec[i];
    // masked ProbSparse self-attention (mix=True)
    gemm(stream, Xb, p.sa.wq, p.sa.bq, Qb, Rd, 512, 512, FB_BIAS);
    gemm(stream, Xb, p.sa.wk, p.sa.bk, Kbuf, Rd, 512, 512, FB_BIAS);
    gemm(stream, Xb, p.sa.wv, p.sa.bv, Vbuf, Rd, 512, 512, FB_BIAS);
    int U = 25, u = 25;  // 5*ceil(ln 144)
    uint32_t fk0, fk1;
    threefry2x32(0u, 1234u, 0u, (uint32_t)(100 + i), &fk0, &fk1);
    k_fill_idx<<<gs(Ld * U), 256, 0, stream>>>(IdxI, Ld * U, Ld, fk0, fk1);
    k_meas<<<1024, Ld, 0, stream>>>(Qb, Kbuf, IdxI, Mb, Ld, U);
    k_topk<<<1024, 32, 0, stream>>>(Mb, TopI, Ld, u);
    k_cumsum<<<gs(BN * 8 * 64), 256, 0, stream>>>(Vbuf, Tb, Ld, 1, BN * 8 * 64);
    k_topattn<<<dim3(1024, (u + 3) / 4), 128, 0, stream>>>(Qb, Kbuf, Vbuf, TopI, Tb,
                                                           Ld, u, 1, 1);
    gemm(stream, Tb, p.sa.wo, p.sa.bo, Yb, Rd, 512, 512, FB_BIAS);
    k_add_ln<<<Rd, 256, 0, stream>>>(Xb, Yb, p.l1g, p.l1b, Xb, Rd);
    // full cross-attention vs encoder memory (mix=False)
    gemm(stream, Xb, p.ca.wq, p.ca.bq, Qb, Rd, 512, 512, FB_BIAS);
    gemm(stream, Eb, p.ca.wk, p.ca.bk, Kbuf, Renc, 512, 512, FB_BIAS);
    gemm(stream, Eb, p.ca.wv, p.ca.bv, Vbuf, Renc, 512, 512, FB_BIAS);
    k_cross<<<(BN * 8 * Ld + 3) / 4, 128, 0, stream>>>(Qb, Kbuf, Vbuf, Tb, Ld, 48);
    gemm(stream, Tb, p.ca.wo, p.ca.bo, Yb, Rd, 512, 512, FB_BIAS);
    k_add_ln<<<Rd, 256, 0, stream>>>(Xb, Yb, p.l2g, p.l2b, Xb, Rd);
    // FFN
    gemm(stream, Xb, p.w1, p.b1, Hb, Rd, 2048, 512, FB_BIAS | FB_GELU);
    gemm(stream, Hb, p.w2, p.b2, Yb, Rd, 512, 2048, FB_BIAS);
    k_add_ln<<<Rd, 256, 0, stream>>>(Xb, Yb, p.l3g, p.l3b, Xb, Rd);
  }
  k_add_ln<<<Rd, 256, 0, stream>>>(Xb, nullptr, dec_ng, dec_nb, Xb, Rd);

  // projection -> mean over 7 channels (linear => fold into averaged weights)
  k_wavg<<<2, 256, 0, stream>>>(proj_w, proj_b, Wavg);
  k_projmean<<<(Rd + 7) / 8, 256, 0, stream>>>(Xb, Wavg, Dm, Rd);
  k_ygather<<<gs(4 * 48 * 32), 256, 0, stream>>>(Dm, Yg, 4 * 48 * 32);

  // final head: (192,32)@(32,512) gelu -> (192,512)@(512,32) -> d_out
  gemm(stream, Yg, fl_w1, fl_b1, Hb, 192, 512, 32, FB_BIAS | FB_GELU);
  gemm(stream, Hb, fl_w2, fl_b2, (float*)d_out, 192, 32, 512, FB_BIAS);
}